// GATConvLayer_68169720922763
// MI455X (gfx1250) — compile-verified
//
#include <hip/hip_runtime.h>
#include <hip/hip_bf16.h>

#define N_NODES 50000
#define N_EDGES 1600000
#define IN_F    128
#define OUT_F   32
#define HEADS   4
#define EDGE_F  4
#define EDGE_HID 32
#define HF      (HEADS * OUT_F)   // 128

typedef __attribute__((ext_vector_type(2))) float v2f;
typedef __attribute__((ext_vector_type(8))) float v8f;

// ---------------------------------------------------------------------------
// K0: init out accumulator to 0, last[] to -1
// ---------------------------------------------------------------------------
__global__ __launch_bounds__(256) void init_kernel(float* __restrict__ out,
                                                   int* __restrict__ last) {
    int i = blockIdx.x * 256 + threadIdx.x;
    if (i < N_NODES * OUT_F) out[i] = 0.0f;
    if (i < N_NODES) last[i] = -1;
}

// ---------------------------------------------------------------------------
// K1: h = x @ W + b   via V_WMMA_F32_16X16X4_F32 (wave32, 16x16 tile/wave)
//   grid.x = ceil(N/16) row tiles, 256 threads = 8 waves, wave w -> col tile w
//   x row-tile (16x128, 8KB) staged in LDS once per workgroup.
// ---------------------------------------------------------------------------
__global__ __launch_bounds__(256) void gemm_h_kernel(const float* __restrict__ x,
                                                     const float* __restrict__ W,
                                                     const float* __restrict__ bias,
                                                     float* __restrict__ h) {
    __shared__ float xs[16 * IN_F];                 // 8 KB
    const int row0 = blockIdx.x * 16;
    const int tid  = threadIdx.x;

    // cooperative stage of x[row0:row0+16, 0:128] -> LDS (ds_store path)
    for (int i = tid; i < 16 * IN_F; i += 256) {
        const int r = i >> 7, c = i & 127;
        const int row = row0 + r;
        xs[i] = (row < N_NODES) ? x[row * IN_F + c] : 0.0f;
    }
    __syncthreads();

    const int wave  = tid >> 5;          // 0..7 -> output column tile
    const int lane  = tid & 31;
    const int col0  = wave * 16;
    const int m     = lane & 15;         // M (A) / N (B,C)
    const int khalf = lane >> 4;         // 0: K=0,1   1: K=2,3

    v8f c = {};
    #pragma unroll
    for (int k0 = 0; k0 < IN_F; k0 += 4) {
        // A 16x4: lanes0-15 VGPR0/1 = K0,K1 ; lanes16-31 = K2,K3
        v2f a, b;
        const int ka = k0 + 2 * khalf;
        a.x = xs[m * IN_F + ka + 0];
        a.y = xs[m * IN_F + ka + 1];
        // B 4x16 (K x N), row-striped across lanes, same K split as A
        b.x = W[(ka + 0) * HF + col0 + m];
        b.y = W[(ka + 1) * HF + col0 + m];
        c = __builtin_amdgcn_wmma_f32_16x16x4_f32(
                /*neg_a=*/false, a, /*neg_b=*/false, b,
                /*c_mod=*/(short)0, c, /*reuse_a=*/false, /*reuse_b=*/false);
    }

    // C/D 16x16: VGPR r -> M = r + 8*khalf, N = m
    const float bv = bias[col0 + m];
    #pragma unroll
    for (int r = 0; r < 8; ++r) {
        const int row = row0 + khalf * 8 + r;
        if (row < N_NODES) h[row * HF + col0 + m] = c[r] + bv;
    }
}

// ---------------------------------------------------------------------------
// K2: per-node attention scores  s[n,h] = <h[n,h,:], att_src[h,:]>  (and d)
//   one wave per node, lane = feature, __shfl_xor tree reduce (wave32)
// ---------------------------------------------------------------------------
__global__ __launch_bounds__(256) void node_scores_kernel(const float* __restrict__ hbuf,
                                                          const float* __restrict__ att_src,
                                                          const float* __restrict__ att_dst,
                                                          float* __restrict__ s,
                                                          float* __restrict__ d) {
    const int node = (blockIdx.x * 256 + threadIdx.x) >> 5;
    const int lane = threadIdx.x & 31;
    if (node >= N_NODES) return;

    const float* hr = hbuf + node * HF;
    float ps[HEADS], pd[HEADS];
    #pragma unroll
    for (int hh = 0; hh < HEADS; ++hh) {
        const float v = hr[hh * OUT_F + lane];
        ps[hh] = v * att_src[hh * OUT_F + lane];
        pd[hh] = v * att_dst[hh * OUT_F + lane];
    }
    #pragma unroll
    for (int off = 16; off > 0; off >>= 1) {
        #pragma unroll
        for (int hh = 0; hh < HEADS; ++hh) {
            ps[hh] += __shfl_xor(ps[hh], off, 32);
            pd[hh] += __shfl_xor(pd[hh], off, 32);
        }
    }
    if (lane == 0) {
        #pragma unroll
        for (int hh = 0; hh < HEADS; ++hh) {
            s[node * HEADS + hh] = ps[hh];
            d[node * HEADS + hh] = pd[hh];
        }
    }
}

// ---------------------------------------------------------------------------
// K3: edge MLP + leaky-relu alpha + segment_max(last edge id per src)
//   MLP weights (292 floats) staged in LDS, one thread per edge
// ---------------------------------------------------------------------------
__global__ __launch_bounds__(256) void edge_kernel(const int* __restrict__ ei,
                                                   const float* __restrict__ edge_attr,
                                                   const float* __restrict__ eW1,
                                                   const float* __restrict__ eb1,
                                                   const float* __restrict__ eW2,
                                                   const float* __restrict__ eb2,
                                                   const float* __restrict__ s,
                                                   const float* __restrict__ dsc,
                                                   float* __restrict__ alpha,
                                                   int* __restrict__ last) {
    __shared__ float w1[EDGE_F * EDGE_HID];   // 128
    __shared__ float b1[EDGE_HID];            // 32
    __shared__ float w2[EDGE_HID * HEADS];    // 128
    __shared__ float b2v[HEADS];              // 4
    const int tid = threadIdx.x;
    if (tid < 128) { w1[tid] = eW1[tid]; w2[tid] = eW2[tid]; }
    if (tid < EDGE_HID) b1[tid] = eb1[tid];
    if (tid < HEADS)    b2v[tid] = eb2[tid];
    __syncthreads();

    const int e = blockIdx.x * 256 + tid;
    if (e >= N_EDGES) return;

    const float ea0 = edge_attr[e * EDGE_F + 0];
    const float ea1 = edge_attr[e * EDGE_F + 1];
    const float ea2 = edge_attr[e * EDGE_F + 2];
    const float ea3 = edge_attr[e * EDGE_F + 3];

    float acc[HEADS] = {b2v[0], b2v[1], b2v[2], b2v[3]};
    #pragma unroll
    for (int j = 0; j < EDGE_HID; ++j) {
        float t = b1[j] + ea0 * w1[j] + ea1 * w1[32 + j]
                        + ea2 * w1[64 + j] + ea3 * w1[96 + j];
        t = fmaxf(t, 0.0f);
        #pragma unroll
        for (int hh = 0; hh < HEADS; ++hh) acc[hh] += t * w2[j * HEADS + hh];
    }

    const int sn = ei[e];
    const int dn = ei[N_EDGES + e];
    #pragma unroll
    for (int hh = 0; hh < HEADS; ++hh) {
        float a = s[sn * HEADS + hh] + dsc[dn * HEADS + hh] + acc[hh];
        a = (a > 0.0f) ? a : 0.2f * a;                 // leaky_relu(0.2)
        alpha[e * HEADS + hh] = a;
    }
    atomicMax(&last[sn], e);                           // last-write-wins scatter
}

// ---------------------------------------------------------------------------
// K4: dense[n,h] = valid ? alpha[last[n],h] : -inf
// ---------------------------------------------------------------------------
__global__ __launch_bounds__(256) void build_dense_kernel(const int* __restrict__ last,
                                                          const float* __restrict__ alpha,
                                                          float* __restrict__ dense) {
    const int n = blockIdx.x * 256 + threadIdx.x;
    if (n >= N_NODES) return;
    const int le = last[n];
    #pragma unroll
    for (int hh = 0; hh < HEADS; ++hh) {
        dense[n * HEADS + hh] = (le >= 0) ? alpha[le * HEADS + hh]
                                          : -__builtin_inff();
    }
}

// ---------------------------------------------------------------------------
// K5: softmax over the NODE axis, one block (1024 thr) per head
// ---------------------------------------------------------------------------
__global__ __launch_bounds__(1024) void softmax_nodes_kernel(float* __restrict__ dense) {
    const int hd   = blockIdx.x;
    const int tid  = threadIdx.x;
    const int lane = tid & 31;
    const int wid  = tid >> 5;          // 0..31 waves
    __shared__ float red[32];
    __shared__ float stat[2];

    // pass 1: max
    float mx = -__builtin_inff();
    for (int n = tid; n < N_NODES; n += 1024)
        mx = fmaxf(mx, dense[n * HEADS + hd]);
    #pragma unroll
    for (int off = 16; off > 0; off >>= 1) mx = fmaxf(mx, __shfl_xor(mx, off, 32));
    if (lane == 0) red[wid] = mx;
    __syncthreads();
    if (wid == 0) {
        float v = red[lane];
        #pragma unroll
        for (int off = 16; off > 0; off >>= 1) v = fmaxf(v, __shfl_xor(v, off, 32));
        if (lane == 0) stat[0] = v;
    }
    __syncthreads();
    mx = stat[0];

    // pass 2: sum of exp
    float sum = 0.0f;
    for (int n = tid; n < N_NODES; n += 1024)
        sum += __expf(dense[n * HEADS + hd] - mx);
    #pragma unroll
    for (int off = 16; off > 0; off >>= 1) sum += __shfl_xor(sum, off, 32);
    __syncthreads();                     // red[] reuse
    if (lane == 0) red[wid] = sum;
    __syncthreads();
    if (wid == 0) {
        float v = red[lane];
        #pragma unroll
        for (int off = 16; off > 0; off >>= 1) v += __shfl_xor(v, off, 32);
        if (lane == 0) stat[1] = v;
    }
    __syncthreads();
    const float inv = 1.0f / stat[1];

    // pass 3: normalize in place
    for (int n = tid; n < N_NODES; n += 1024) {
        const float v = dense[n * HEADS + hd];
        dense[n * HEADS + hd] = __expf(v - mx) * inv;
    }
}

// ---------------------------------------------------------------------------
// K6: aggregation with head-mean folded in:
//   out[src,f] += 0.25 * sum_h h[dst,h,f] * dense[src,h]
//   thread = (edge, feature); h and dense are L2-resident (192MB L2)
// ---------------------------------------------------------------------------
__global__ __launch_bounds__(256) void aggregate_kernel(const int* __restrict__ ei,
                                                        const float* __restrict__ hbuf,
                                                        const float* __restrict__ dense,
                                                        float* __restrict__ out) {
    const int gid = blockIdx.x * 256 + threadIdx.x;
    const int e = gid >> 5;
    const int f = gid & 31;
    if (e >= N_EDGES) return;
    const int sn = ei[e];
    const int dn = ei[N_EDGES + e];
    const float* hr = hbuf + dn * HF;
    const float* al = dense + sn * HEADS;
    const float acc = hr[0 * OUT_F + f] * al[0]
                    + hr[1 * OUT_F + f] * al[1]
                    + hr[2 * OUT_F + f] * al[2]
                    + hr[3 * OUT_F + f] * al[3];
    atomicAdd(&out[sn * OUT_F + f], 0.25f * acc);
}

// ---------------------------------------------------------------------------
extern "C" void kernel_launch(void* const* d_in, const int* in_sizes, int n_in,
                              void* d_out, int out_size, void* d_ws, size_t ws_size,
                              hipStream_t stream) {
    const float* x        = (const float*)d_in[0];
    const int*   ei       = (const int*)  d_in[1];
    const float* edge_attr= (const float*)d_in[2];
    const float* W        = (const float*)d_in[3];
    const float* bias     = (const float*)d_in[4];
    const float* eW1      = (const float*)d_in[5];
    const float* eb1      = (const float*)d_in[6];
    const float* eW2      = (const float*)d_in[7];
    const float* eb2      = (const float*)d_in[8];
    const float* att_src  = (const float*)d_in[9];
    const float* att_dst  = (const float*)d_in[10];
    float* out = (float*)d_out;

    // workspace carve-up (~53.8 MB)
    float* ws    = (float*)d_ws;
    float* hbuf  = ws;                                  // N*128
    float* s     = hbuf  + (size_t)N_NODES * HF;        // N*4
    float* dsc   = s     + (size_t)N_NODES * HEADS;     // N*4
    float* alpha = dsc   + (size_t)N_NODES * HEADS;     // E*4
    int*   last  = (int*)(alpha + (size_t)N_EDGES * HEADS);  // N
    float* dense = (float*)(last + N_NODES);            // N*4

    const int outElems = N_NODES * OUT_F;
    init_kernel<<<(outElems + 255) / 256, 256, 0, stream>>>(out, last);

    gemm_h_kernel<<<(N_NODES + 15) / 16, 256, 0, stream>>>(x, W, bias, hbuf);

    node_scores_kernel<<<(N_NODES * 32 + 255) / 256, 256, 0, stream>>>(
        hbuf, att_src, att_dst, s, dsc);

    edge_kernel<<<(N_EDGES + 255) / 256, 256, 0, stream>>>(
        ei, edge_attr, eW1, eb1, eW2, eb2, s, dsc, alpha, last);

    build_dense_kernel<<<(N_NODES + 255) / 256, 256, 0, stream>>>(last, alpha, dense);

    softmax_nodes_kernel<<<HEADS, 1024, 0, stream>>>(dense);

    const long aggThreads = (long)N_EDGES * 32;
    aggregate_kernel<<<(int)((aggThreads + 255) / 256), 256, 0, stream>>>(
        ei, hbuf, dense, out);
}